// kernel_generated_2_21887153341350
// MI455X (gfx1250) — compile-verified
//
#include <hip/hip_runtime.h>

typedef __attribute__((ext_vector_type(2))) float v2f;
typedef __attribute__((ext_vector_type(8))) float v8f;

// B-operand gather for the conv GEMM, now from LDS.
// B[(j,i,k),(p,n)] = t4_lds[p, j, n+i+k-2] masked by the COMPOSED unfold padding:
// zero unless 0 <= n+k-1 < 14 (first unfold) AND 0 <= n+i+k-2 < 14 (second).
__device__ __forceinline__ float conv_bval(const float* __restrict__ s_t4,
                                           int lbase, int n, int r) {
  const int j  = r / 9;            // r = j*9 + i*3 + k, r in [0,72)
  const int q  = r - j * 9;
  const int i  = q / 3;
  const int k  = q - i * 3;
  const int s1 = n + k - 1;        // first-unfold source index
  const int s2 = s1 + i - 1;       // final source index
  const bool ok = ((unsigned)s1 < 14u) & ((unsigned)s2 < 14u);
  return ok ? s_t4[lbase + j * 14 + s2] : 0.0f;
}

// ---------------------------------------------------------------------------
// One fused kernel, grid = 14 blocks (one per H row 'o'), 256 threads (8 waves).
// The unfolds act only on W, so each o-slice is fully independent:
//   Phase 1: t4[:, :, o, :]  (16 x 112)  = w_mix (16x64) @ x[:, :, o, :] (64x112)
//            -> LDS, 7 N-tiles, fp32 WMMA 16x16x4.
//   Phase 2: out[l, p, o', n] = sum_{j,i,k} w_conv[l,(j,i,k)] * B  with B from LDS,
//            M=32 (2 tiles), K=72 (18 steps), N=224 (14 tiles), o' = (o+1)%14.
// ---------------------------------------------------------------------------
__global__ __launch_bounds__(256) void fused_mix_unfold_conv(
    const float* __restrict__ x,        // [64][8][14][14] view of (1,512,14,14)
    const float* __restrict__ w_conv,   // [32][72]
    const float* __restrict__ w_mix,    // [16][64]
    float* __restrict__ y) {            // [512][14][14]
  __shared__ float s_mix[16 * 64];      //  4 KB
  __shared__ float s_cv [32 * 72];      //  9 KB
  __shared__ float s_t4 [16 * 112];     //  7 KB: [p][l8*14 + n] for this o

  const int tid  = threadIdx.x;
  const int lane = tid & 31;
  const int wave = tid >> 5;
  const int o    = blockIdx.x;          // 0..13

  // Stage both weight matrices into LDS.
  for (int i = tid; i < 16 * 64; i += 256) s_mix[i] = w_mix[i];
  for (int i = tid; i < 32 * 72; i += 256) s_cv[i]  = w_conv[i];
  __syncthreads();

  const int row   = lane & 15;          // A-matrix M row
  const int kl    = (lane >> 4) * 2;    // per-lane-half local K base (0 or 2)
  const int mbase = (lane >> 4) * 8;    // C-matrix M base for this lane half

  // ---- Phase 1: channel-mix GEMM into LDS (7 tiles, one per wave) ----------
  if (wave < 7) {                       // wave-uniform guard: EXEC stays full
    const int col = wave * 16 + (lane & 15);        // 0..111 = (l8, n)
    const int l8  = col / 14;
    const int n   = col - l8 * 14;
    const size_t gb = (size_t)l8 * 196 + o * 14 + n;
    v8f c = {};
#pragma unroll
    for (int ks = 0; ks < 16; ++ks) {
      const int k0 = ks * 4 + kl;
      v2f a, b;
      a.x = s_mix[row * 64 + k0];                   // A 16x4 fragment
      a.y = s_mix[row * 64 + k0 + 1];
      b.x = x[(size_t)k0 * 1568 + gb];              // B 4x16 fragment (global)
      b.y = x[(size_t)(k0 + 1) * 1568 + gb];
      c = __builtin_amdgcn_wmma_f32_16x16x4_f32(false, a, false, b,
                                                (short)0, c, false, false);
    }
#pragma unroll
    for (int v = 0; v < 8; ++v)                     // C: M=mbase+v, N=col
      s_t4[(mbase + v) * 112 + col] = c[v];
  }
  __syncthreads();

  // ---- Phase 2: unfold^2 + contraction as GEMM with LDS-gathered B ---------
  for (int tile = wave; tile < 14; tile += 8) {     // wave-uniform trip count
    const int col = tile * 16 + (lane & 15);        // 0..223 = (p, n)
    const int p   = col / 14;
    const int n   = col - p * 14;
    const int lbase = p * 112;
    v8f c0 = {}, c1 = {};
#pragma unroll
    for (int ks = 0; ks < 18; ++ks) {               // K = 72 = 18 * 4 exactly
      const int r0 = ks * 4 + kl;
      v2f a0, a1, b;
      a0.x = s_cv[row * 72 + r0];                   // l = row       (M tile 0)
      a0.y = s_cv[row * 72 + r0 + 1];
      a1.x = s_cv[(row + 16) * 72 + r0];            // l = row + 16  (M tile 1)
      a1.y = s_cv[(row + 16) * 72 + r0 + 1];
      b.x  = conv_bval(s_t4, lbase, n, r0);         // shared B, LDS gather
      b.y  = conv_bval(s_t4, lbase, n, r0 + 1);
      c0 = __builtin_amdgcn_wmma_f32_16x16x4_f32(false, a0, false, b,
                                                 (short)0, c0, false, false);
      c1 = __builtin_amdgcn_wmma_f32_16x16x4_f32(false, a1, false, b,
                                                 (short)0, c1, false, false);
    }
    // out channel = l*16 + p ; out H row = (o+1)%14 (the roll) ; out W col = n
    const int oo    = (o + 1 == 14) ? 0 : (o + 1);
    const int obase = p * 196 + oo * 14 + n;
#pragma unroll
    for (int v = 0; v < 8; ++v) {
      const int l0 = mbase + v;
      y[(size_t)l0 * 3136 + obase]        = c0[v];  // (l0*16 + p)*196 + ...
      y[(size_t)(l0 + 16) * 3136 + obase] = c1[v];
    }
  }
}

// ---------------------------------------------------------------------------
extern "C" void kernel_launch(void* const* d_in, const int* in_sizes, int n_in,
                              void* d_out, int out_size, void* d_ws, size_t ws_size,
                              hipStream_t stream) {
  const float* x      = (const float*)d_in[0];   // (1,512,14,14)
  const float* w_conv = (const float*)d_in[1];   // (32,8,3,3)
  const float* w_mix  = (const float*)d_in[2];   // (16,64)
  float* y = (float*)d_out;                      // (1,512,14,14)

  fused_mix_unfold_conv<<<14, 256, 0, stream>>>(x, w_conv, w_mix, y);
}